// ContinuousLSCSS_45956150068052
// MI455X (gfx1250) — compile-verified
//
#include <hip/hip_runtime.h>
#include <hip/hip_bf16.h>
#include <math.h>

typedef __attribute__((ext_vector_type(2))) float v2f;
typedef __attribute__((ext_vector_type(8))) float v8f;

#define DCOL 4096          // d (columns of A)
#define NROW 4096          // n (rows of A)
#define KSEL 8             // k
#define CAP  16            // 2*k
#define EPSF 1e-10f
#define NT   4             // N-tiles per wave in the WMMA kernel

// ---------------------------------------------------------------------------
// hash RNG -> uniform (0,1); deterministic in (step, j)
// ---------------------------------------------------------------------------
__device__ __forceinline__ float u01(uint32_t s, uint32_t j) {
    uint32_t x = s * 0x9E3779B9u ^ (j + 0x7F4A7C15u) * 0x85EBCA6Bu;
    x ^= x >> 16; x *= 0x7FEB352Du;
    x ^= x >> 15; x *= 0x846CA68Bu;
    x ^= x >> 16;
    return (float)(x >> 8) * (1.0f / 16777216.0f);
}

// ---------------------------------------------------------------------------
// init: I_soft = 0, diag_val = 0
// ---------------------------------------------------------------------------
__global__ void init_kernel(float* I_soft, float* scal) {
    int j = blockIdx.x * blockDim.x + threadIdx.x;
    if (j < DCOL) I_soft[j] = 0.0f;
    if (j == 0) scal[0] = 0.0f;
}

// ---------------------------------------------------------------------------
// column norms of (A + dv*I): colA2[j] = sum_i (A[i,j] + (i==j)dv)^2
// ---------------------------------------------------------------------------
__global__ void colnorms_kernel(const float* __restrict__ A,
                                const float* __restrict__ scal,
                                float* __restrict__ colA2) {
    int j = blockIdx.x * blockDim.x + threadIdx.x;
    if (j >= DCOL) return;
    float dv = scal[0];
    float acc = 0.0f;
    for (int i = 0; i < NROW; ++i) {
        float a = A[(size_t)i * DCOL + j];
        if (i == j) a += dv;
        acc = fmaf(a, a, acc);
    }
    colA2[j] = acc;
}

// ---------------------------------------------------------------------------
// capped select: indices with I_soft > 0.5 (at most CAP by construction)
// ---------------------------------------------------------------------------
__global__ void gather_selected_kernel(const float* __restrict__ I_soft,
                                       int* __restrict__ idx,
                                       float* __restrict__ w) {
    if (threadIdx.x == 0) {
        int c = 0;
        for (int j = 0; j < DCOL && c < CAP; ++j) {
            if (I_soft[j] > 0.5f) { idx[c] = j; w[c] = 1.0f; ++c; }
        }
        for (; c < CAP; ++c) { idx[c] = 0; w[c] = 0.0f; }
    }
}

// ---------------------------------------------------------------------------
// WMMA fp32 GEMM (pure):  Y (16 x DCOL) = A[:, idx]^T  x  A
// No dv/w folding in the hot loop -- applied algebraically in y_correct.
// Each wave owns NT=4 adjacent 16x16 output tiles (4 accumulators), so the
// gathered A-operand (selected columns) is loaded once per K-step and reused
// for 4 chained V_WMMA_F32_16X16X4_F32.
//
// A-operand (16x4, MxK): lanes 0-15 M=0..15; VGPR0 = K 0/2 per lane half,
// VGPR1 = K 1/3.  B-operand (4x16, KxN) mirrored.  C/D (16x16 f32): VGPR r
// holds M=r (lanes 0-15) / M=8+r (lanes 16-31), N = lane&15.
// blockDim = 128 (4 waves); grid.x = DCOL/16/NT/4 = 16; grid.y = R/16.
// ---------------------------------------------------------------------------
__global__ void wmma_gram_kernel(const float* __restrict__ A,
                                 const int* __restrict__ idx,
                                 float* __restrict__ Y) {
    const int wave   = threadIdx.x >> 5;
    const int lane   = threadIdx.x & 31;
    const int m      = lane & 15;
    const int khalf  = lane >> 4;
    const int mtile  = blockIdx.y;
    const int ntile0 = (blockIdx.x * 4 + wave) * NT;
    const int n0     = ntile0 * 16;

    const int colm = idx[mtile * 16 + m];
    const int cb   = n0 + m;

    const v8f zero = {0.f, 0.f, 0.f, 0.f, 0.f, 0.f, 0.f, 0.f};
    v8f acc[NT];
    #pragma unroll
    for (int tN = 0; tN < NT; ++tN) acc[tN] = zero;

    for (int k = 0; k < NROW; k += 4) {
        const size_t o0 = (size_t)(k + 2 * khalf) * DCOL;
        const size_t o1 = o0 + DCOL;
        v2f av;
        av.x = A[o0 + colm];
        av.y = A[o1 + colm];
        #pragma unroll
        for (int tN = 0; tN < NT; ++tN) {
            v2f bv;
            bv.x = A[o0 + cb + 16 * tN];
            bv.y = A[o1 + cb + 16 * tN];
            acc[tN] = __builtin_amdgcn_wmma_f32_16x16x4_f32(
                          false, av, false, bv, (short)0, acc[tN], false, false);
        }
    }

    #pragma unroll
    for (int tN = 0; tN < NT; ++tN) {
        #pragma unroll
        for (int r = 0; r < 8; ++r) {
            const int row = mtile * 16 + r + 8 * khalf;
            Y[(size_t)row * DCOL + n0 + 16 * tN + m] = acc[tN][r];
        }
    }
}

// ---------------------------------------------------------------------------
// epilogue: fold w and diag_val into Y (in place):
// Y[p,j] = w_p * ( Y~[p,j] + dv*(A[idx_p,j] + A[j,idx_p]) + dv^2*(j==idx_p) )
// grid = (DCOL/256, 16)
// ---------------------------------------------------------------------------
__global__ void y_correct_kernel(const float* __restrict__ A,
                                 const float* __restrict__ scal,
                                 const int* __restrict__ idx,
                                 const float* __restrict__ w,
                                 float* __restrict__ Y) {
    int j = blockIdx.x * blockDim.x + threadIdx.x;
    int p = blockIdx.y;
    if (j >= DCOL) return;
    const float dv = scal[0];
    const int ip = idx[p];
    float corr = dv * (A[(size_t)ip * DCOL + j] + A[(size_t)j * DCOL + ip]);
    if (j == ip) corr += dv * dv;
    Y[(size_t)p * DCOL + j] = w[p] * (Y[(size_t)p * DCOL + j] + corr);
}

// ---------------------------------------------------------------------------
// tiny Gram: G[p][q] = Y[p, idx[q]] * w[q]   (Y rows already carry w[p])
// ---------------------------------------------------------------------------
__global__ void gram_small_kernel(const float* __restrict__ Y,
                                  const int* __restrict__ idx,
                                  const float* __restrict__ w,
                                  float* __restrict__ G) {
    int t = threadIdx.x;
    if (t < CAP * CAP) {
        int p = t / CAP, q = t % CAP;
        G[t] = Y[(size_t)p * DCOL + idx[q]] * w[q];
    }
}

// ---------------------------------------------------------------------------
// 16x16 Gauss-Jordan pseudo-inverse (pivot-skip on near-zero pivots).
// blockDim = 32 (t indexes augmented column).
// ---------------------------------------------------------------------------
__global__ void pinv16_kernel(const float* __restrict__ G,
                              float* __restrict__ H) {
    __shared__ float Mx[CAP][2 * CAP];
    __shared__ float fcol[CAP];
    __shared__ float piv;
    __shared__ float tol;
    const int t = threadIdx.x; // 0..31
    for (int p = 0; p < CAP; ++p)
        Mx[p][t] = (t < CAP) ? G[p * CAP + t] : ((t - CAP) == p ? 1.0f : 0.0f);
    __syncthreads();
    if (t == 0) {
        float tr = 0.0f;
        for (int p = 0; p < CAP; ++p) tr += fabsf(Mx[p][p]);
        tol = tr * 1e-7f + 1e-30f;
    }
    __syncthreads();
    for (int p = 0; p < CAP; ++p) {
        if (t == 0) piv = Mx[p][p];
        __syncthreads();
        if (fabsf(piv) > tol) {
            Mx[p][t] = Mx[p][t] / piv;
            __syncthreads();
            if (t < CAP) fcol[t] = (t == p) ? 0.0f : Mx[t][p];
            __syncthreads();
            for (int r = 0; r < CAP; ++r)
                Mx[r][t] -= fcol[r] * Mx[p][t];
            __syncthreads();
        } else {
            Mx[p][t] = 0.0f; // singular direction -> zero row (pinv-like)
            __syncthreads();
        }
    }
    if (t < CAP)
        for (int p = 0; p < CAP; ++p) H[p * CAP + t] = Mx[p][CAP + t];
}

// ---------------------------------------------------------------------------
// residual column norms via projector quadratic form:
// cn2[j] = max(0, ||A_j||^2 - y_j^T H y_j),  y_j = Y[:, j]
// ---------------------------------------------------------------------------
__global__ void quadform_kernel(const float* __restrict__ Y,
                                const float* __restrict__ H,
                                const float* __restrict__ colA2,
                                float* __restrict__ cn2) {
    int j = blockIdx.x * blockDim.x + threadIdx.x;
    if (j >= DCOL) return;
    float y[CAP];
    #pragma unroll
    for (int p = 0; p < CAP; ++p) y[p] = Y[(size_t)p * DCOL + j];
    float q = 0.0f;
    #pragma unroll
    for (int p = 0; p < CAP; ++p) {
        float hy = 0.0f;
        #pragma unroll
        for (int c = 0; c < CAP; ++c) hy = fmaf(H[p * CAP + c], y[c], hy);
        q = fmaf(y[p], hy, q);
    }
    float v = colA2[j] - q;
    cn2[j] = (v > 0.0f) ? v : 0.0f;
}

// ---------------------------------------------------------------------------
// Gumbel-softmax over all d columns + I_soft update.  Single block, 1024 thr.
// ---------------------------------------------------------------------------
__global__ void select_update_kernel(const float* __restrict__ cn2,
                                     float* __restrict__ I_soft,
                                     int step) {
    __shared__ float red[1024];
    const int t = threadIdx.x;
    float s = 0.0f;
    for (int j = t; j < DCOL; j += 1024) s += cn2[j];
    red[t] = s; __syncthreads();
    for (int o = 512; o > 0; o >>= 1) { if (t < o) red[t] += red[t + o]; __syncthreads(); }
    const float total = red[0] + EPSF;
    __syncthreads();

    float logit[4];
    float lm = -1e30f;
    #pragma unroll
    for (int c = 0; c < 4; ++c) {
        int j = t + c * 1024;
        float prob = cn2[j] / total;
        float u = u01((uint32_t)(step + 1) * 0x68BC21EBu, (uint32_t)j);
        float g = -logf(-logf(u + EPSF) + EPSF);
        logit[c] = logf(prob + EPSF) + g; // TEMPERATURE = 1
        lm = fmaxf(lm, logit[c]);
    }
    red[t] = lm; __syncthreads();
    for (int o = 512; o > 0; o >>= 1) { if (t < o) red[t] = fmaxf(red[t], red[t + o]); __syncthreads(); }
    const float mx = red[0];
    __syncthreads();

    float ex[4], es = 0.0f;
    #pragma unroll
    for (int c = 0; c < 4; ++c) { ex[c] = expf(logit[c] - mx); es += ex[c]; }
    red[t] = es; __syncthreads();
    for (int o = 512; o > 0; o >>= 1) { if (t < o) red[t] += red[t + o]; __syncthreads(); }
    const float Z = red[0];

    #pragma unroll
    for (int c = 0; c < 4; ++c) {
        int j = t + c * 1024;
        float soft = ex[c] / Z;
        float old = I_soft[j];
        I_soft[j] = old + soft * (1.0f - old);
    }
}

// ---------------------------------------------------------------------------
// end of phase t=0: diag_val = ||E0||_F / ((52*min(n,d)*(k+1))^1.5 + eps);
// reset I_soft.
// ---------------------------------------------------------------------------
__global__ void dv_kernel(const float* __restrict__ cn2,
                          float* __restrict__ scal,
                          float* __restrict__ I_soft) {
    __shared__ float red[1024];
    const int t = threadIdx.x;
    float s = 0.0f;
    for (int j = t; j < DCOL; j += 1024) s += cn2[j];
    red[t] = s; __syncthreads();
    for (int o = 512; o > 0; o >>= 1) { if (t < o) red[t] += red[t + o]; __syncthreads(); }
    if (t == 0) {
        float nf = sqrtf(red[0]);
        float base = 52.0f * (float)DCOL * (float)(KSEL + 1);
        scal[0] = nf / (powf(base, 1.5f) + EPSF);
    }
    for (int j = t; j < DCOL; j += 1024) I_soft[j] = 0.0f;
}

// ---------------------------------------------------------------------------
// final top-k over score (selected columns ranked ascending, then I_soft)
// ---------------------------------------------------------------------------
__global__ void final_select_kernel(const float* __restrict__ I_soft,
                                    int* __restrict__ I_final) {
    if (threadIdx.x == 0) {
        for (int c = 0; c < KSEL; ++c) {
            float best = -1e30f; int bi = 0;
            for (int j = 0; j < DCOL; ++j) {
                bool used = false;
                for (int q = 0; q < c; ++q) if (I_final[q] == j) used = true;
                if (used) continue;
                float Is = I_soft[j];
                float sc = (Is > 0.5f) ? (2.0f + (float)(DCOL - j) / (float)DCOL) : Is;
                if (sc > best) { best = sc; bi = j; }
            }
            I_final[c] = bi;
        }
    }
}

// ---------------------------------------------------------------------------
// output: [ A_prime[:, I_final] (n x k, row-major) | I_final as floats ]
// ---------------------------------------------------------------------------
__global__ void output_kernel(const float* __restrict__ A,
                              const float* __restrict__ scal,
                              const int* __restrict__ I_final,
                              float* __restrict__ out) {
    int tid = blockIdx.x * blockDim.x + threadIdx.x;
    const int total = NROW * KSEL;
    const float dv = scal[0];
    if (tid < total) {
        int i = tid / KSEL, c = tid % KSEL;
        int col = I_final[c];
        float v = A[(size_t)i * DCOL + col];
        if (i == col) v += dv;
        out[tid] = v;
    }
    if (tid < KSEL) out[total + tid] = (float)I_final[tid];
}

// ---------------------------------------------------------------------------
extern "C" void kernel_launch(void* const* d_in, const int* in_sizes, int n_in,
                              void* d_out, int out_size, void* d_ws, size_t ws_size,
                              hipStream_t stream) {
    (void)in_sizes; (void)n_in; (void)out_size; (void)ws_size;
    const float* A = (const float*)d_in[0];
    float* out = (float*)d_out;

    // workspace carve (~320 KB)
    float* ws     = (float*)d_ws;
    float* colA2  = ws;                 // DCOL
    float* Y      = colA2 + DCOL;       // 16*DCOL
    float* G      = Y + 16 * DCOL;      // 256
    float* H      = G + 256;            // 256
    float* cn2    = H + 256;            // DCOL
    float* I_soft = cn2 + DCOL;         // DCOL
    float* scal   = I_soft + DCOL;      // 4
    float* wsel   = scal + 4;           // 16
    int*   idx    = (int*)(wsel + 16);  // 16
    int*   I_fin  = idx + 16;           // 8

    init_kernel<<<16, 256, 0, stream>>>(I_soft, scal);
    colnorms_kernel<<<16, 256, 0, stream>>>(A, scal, colA2);

    int step = 0;
    for (int t = 0; t < 2; ++t) {
        for (int it = 0; it < KSEL; ++it) {
            gather_selected_kernel<<<1, 32, 0, stream>>>(I_soft, idx, wsel);
            // pure Gram via WMMA (no fold logic in hot loop)
            wmma_gram_kernel<<<dim3(DCOL / 16 / NT / 4, 1), 128, 0, stream>>>(A, idx, Y);
            // fold w and diag_val algebraically
            y_correct_kernel<<<dim3(DCOL / 256, CAP), 256, 0, stream>>>(A, scal, idx, wsel, Y);
            gram_small_kernel<<<1, 256, 0, stream>>>(Y, idx, wsel, G);
            pinv16_kernel<<<1, 32, 0, stream>>>(G, H);
            quadform_kernel<<<16, 256, 0, stream>>>(Y, H, colA2, cn2);
            if (t == 0 && it == KSEL - 1) {
                // E0 = residual of this step; derive diag_val, reset I_soft,
                // refresh column norms for A + dv*I used in phase t=1.
                dv_kernel<<<1, 1024, 0, stream>>>(cn2, scal, I_soft);
                colnorms_kernel<<<16, 256, 0, stream>>>(A, scal, colA2);
            } else {
                select_update_kernel<<<1, 1024, 0, stream>>>(cn2, I_soft, step);
            }
            ++step;
        }
    }
    // ContinuousLS refinement loop is dead code (result discarded) -> skipped.
    final_select_kernel<<<1, 32, 0, stream>>>(I_soft, I_fin);
    output_kernel<<<(NROW * KSEL + 255) / 256, 256, 0, stream>>>(A, scal, I_fin, out);
}